// RedditSkip_63848983822724
// MI455X (gfx1250) — compile-verified
//
#include <hip/hip_runtime.h>

typedef __attribute__((ext_vector_type(2))) float v2f;
typedef __attribute__((ext_vector_type(8))) float v8f;

#define NNODES 100000
#define NEDGES 3200000

static __device__ __forceinline__ v8f wmma_k4(v2f a, v2f b, v8f c) {
  // D = A(16x4 f32) * B(4x16 f32) + C(16x16 f32)
  return __builtin_amdgcn_wmma_f32_16x16x4_f32(false, a, false, b, (short)0, c,
                                               false, false);
}

// async copy of 16 bytes global -> LDS (ASYNCcnt-tracked, no VGPR round trip)
static __device__ __forceinline__ void async_ld16(unsigned lds_byte,
                                                  unsigned goff_byte,
                                                  const void* base) {
  asm volatile("global_load_async_to_lds_b128 %0, %1, %2"
               :
               : "v"(lds_byte), "v"(goff_byte), "s"(base)
               : "memory");
}

static __device__ __forceinline__ void wait_async0() {
#if __has_builtin(__builtin_amdgcn_s_wait_asynccnt)
  __builtin_amdgcn_s_wait_asynccnt(0);
#else
  asm volatile("s_wait_asynccnt 0x0" ::: "memory");
#endif
}

// ---------------- degree / edge prep ----------------
__global__ void k_deg_init(float* __restrict__ deg) {
  int i = blockIdx.x * blockDim.x + threadIdx.x;
  if (i < NNODES) deg[i] = 1.0f;  // self loop
}

// degree atomics + one-time int64 -> int32 edge conversion
__global__ void k_prep_edges(const long long* __restrict__ ei,
                             float* __restrict__ deg, int* __restrict__ s32,
                             int* __restrict__ d32) {
  int e = blockIdx.x * blockDim.x + threadIdx.x;
  if (e < NEDGES) {
    int s = (int)ei[e];
    int d = (int)ei[NEDGES + e];
    s32[e] = s;
    d32[e] = d;
    atomicAdd(&deg[d], 1.0f);
  }
}

__global__ void k_dis(float* __restrict__ deg) {
  int i = blockIdx.x * blockDim.x + threadIdx.x;
  if (i < NNODES) deg[i] = rsqrtf(deg[i]);
}

// per-edge normalization precomputed once (reused by all 3 GCN layers)
__global__ void k_norm(const int* __restrict__ s32, const int* __restrict__ d32,
                       const float* __restrict__ dis,
                       float* __restrict__ norm) {
  int e = blockIdx.x * blockDim.x + threadIdx.x;
  if (e < NEDGES) norm[e] = dis[s32[e]] * dis[d32[e]];
}

// ---------------- RW = R @ We1[128:192,:]  (256x64) ----------------
__global__ void k_rw(const float* __restrict__ R, const float* __restrict__ We1,
                     float* __restrict__ RW) {
  int idx = blockIdx.x * blockDim.x + threadIdx.x;
  if (idx >= 256 * 64) return;
  int i = idx >> 6, j = idx & 63;
  float s = 0.f;
#pragma unroll 8
  for (int k = 0; k < 64; ++k) s += R[i * 64 + k] * We1[(128 + k) * 64 + j];
  RW[idx] = s;
}

// ---------------- fused embed MLP: h0 = tanh(tanh([x|S@R]@We1+be1)@We2+be2) --
// block = 128 threads (4 waves), 64 rows; double-buffered async LDS staging
__global__ __launch_bounds__(128) void k_embed(
    const float* __restrict__ x, const float* __restrict__ S,
    const float* __restrict__ We1, const float* __restrict__ be1,
    const float* __restrict__ We2, const float* __restrict__ be2,
    const float* __restrict__ RW, float* __restrict__ h0) {
  __shared__ float Abuf[2][64 * 132];  // 2 x staged 64x128 chunk, padded
  __shared__ float T1[64 * 68];        // tanh(layer-1) intermediate, padded
  const int tid = threadIdx.x;
  const int lane = tid & 31;
  const int wave = tid >> 5;
  const int rowBase = blockIdx.x * 64;
  const int lrb = wave * 16;
  const int n = lane & 15;
  const int kk = (lane >> 4) << 1;  // lanes 16-31 hold K+2,K+3
  const int hi = (lane >> 4) * 8;

  auto stage = [&](int phase, int bufIdx) {
    const float* src;
    int ldK, koff;
    if (phase == 0)      { src = x; ldK = 128; koff = 0; }
    else if (phase == 1) { src = S; ldK = 256; koff = 0; }
    else                 { src = S; ldK = 256; koff = 128; }
    float* dstb = Abuf[bufIdx];
#pragma unroll
    for (int i = tid; i < 64 * 32; i += 128) {
      int r = i >> 5, c4 = i & 31;
      int gr = rowBase + r;
      if (gr < NNODES) {
        unsigned lo = (unsigned)(size_t)(&dstb[r * 132 + c4 * 4]);
        unsigned go = (unsigned)(((size_t)gr * ldK + koff + c4 * 4) * 4);
        async_ld16(lo, go, src);
      }
      // OOB rows keep stale LDS data: only feeds discarded output rows
    }
  };

  v8f zero = {};
  v8f acc[4];
  acc[0] = zero; acc[1] = zero; acc[2] = zero; acc[3] = zero;

  stage(0, 0);
  wait_async0();
  __syncthreads();

  int cur = 0;
  for (int phase = 0; phase < 3; ++phase) {
    if (phase < 2) stage(phase + 1, cur ^ 1);  // prefetch next chunk (async)
    const float* Bw = (phase == 0) ? We1 : ((phase == 1) ? RW : RW + 128 * 64);
    const float* Ab = Abuf[cur];
#pragma unroll 4
    for (int k0 = 0; k0 < 128; k0 += 4) {
      v2f a = *(const v2f*)(&Ab[(lrb + n) * 132 + k0 + kk]);
#pragma unroll
      for (int t = 0; t < 4; ++t) {
        v2f b;
        b.x = Bw[(k0 + kk) * 64 + t * 16 + n];
        b.y = Bw[(k0 + kk + 1) * 64 + t * 16 + n];
        acc[t] = wmma_k4(a, b, acc[t]);
      }
    }
    wait_async0();
    __syncthreads();
    cur ^= 1;
  }

  // bias + tanh -> T1 (each wave only touches its own 16 rows)
#pragma unroll
  for (int t = 0; t < 4; ++t) {
    float bias = be1[t * 16 + n];
#pragma unroll
    for (int v = 0; v < 8; ++v)
      T1[(lrb + v + hi) * 68 + t * 16 + n] = tanhf(acc[t][v] + bias);
  }

  // GEMM2: T1(16x64) @ We2(64x32)
  v8f acc2[2];
  acc2[0] = zero; acc2[1] = zero;
#pragma unroll 4
  for (int k0 = 0; k0 < 64; k0 += 4) {
    v2f a = *(const v2f*)(&T1[(lrb + n) * 68 + k0 + kk]);
#pragma unroll
    for (int t = 0; t < 2; ++t) {
      v2f b;
      b.x = We2[(k0 + kk) * 32 + t * 16 + n];
      b.y = We2[(k0 + kk + 1) * 32 + t * 16 + n];
      acc2[t] = wmma_k4(a, b, acc2[t]);
    }
  }
#pragma unroll
  for (int t = 0; t < 2; ++t) {
    float bias = be2[t * 16 + n];
#pragma unroll
    for (int v = 0; v < 8; ++v) {
      int gr = rowBase + lrb + v + hi;
      if (gr < NNODES)
        h0[(size_t)gr * 32 + t * 16 + n] = tanhf(acc2[t][v] + bias);
    }
  }
}

// ---------------- 32x32 WMMA GEMM -------------------------------------------
// MODE 0: out0 = hin@W  (xw), out1 = xw * dis^2 (agg init with self-loop)
// MODE 1: out0 = tanh(hin@W + bias)
template <int MODE>
__global__ __launch_bounds__(128) void k_gemm32(
    const float* __restrict__ hin, const float* __restrict__ W,
    const float* __restrict__ bias, const float* __restrict__ dis,
    float* __restrict__ out0, float* __restrict__ out1) {
  __shared__ float Als[64 * 36];
  const int tid = threadIdx.x;
  const int lane = tid & 31;
  const int wave = tid >> 5;
  const int rowBase = blockIdx.x * 64;
  const int lrb = wave * 16;
  const int n = lane & 15;
  const int kk = (lane >> 4) << 1;
  const int hi = (lane >> 4) * 8;

#pragma unroll
  for (int i = tid; i < 64 * 8; i += 128) {
    int r = i >> 3, c4 = i & 7;
    int gr = rowBase + r;
    if (gr < NNODES) {
      unsigned lo = (unsigned)(size_t)(&Als[r * 36 + c4 * 4]);
      unsigned go = (unsigned)(((size_t)gr * 32 + c4 * 4) * 4);
      async_ld16(lo, go, hin);
    }
  }
  wait_async0();
  __syncthreads();

  v8f zero = {};
  v8f acc[2];
  acc[0] = zero; acc[1] = zero;
#pragma unroll
  for (int k0 = 0; k0 < 32; k0 += 4) {
    v2f a = *(const v2f*)(&Als[(lrb + n) * 36 + k0 + kk]);
#pragma unroll
    for (int t = 0; t < 2; ++t) {
      v2f b;
      b.x = W[(k0 + kk) * 32 + t * 16 + n];
      b.y = W[(k0 + kk + 1) * 32 + t * 16 + n];
      acc[t] = wmma_k4(a, b, acc[t]);
    }
  }

  float dd[8];
  if (MODE == 0) {
#pragma unroll
    for (int v = 0; v < 8; ++v) {
      int gr = rowBase + lrb + v + hi;
      float dv = (gr < NNODES) ? dis[gr] : 0.f;
      dd[v] = dv * dv;
    }
  }
#pragma unroll
  for (int t = 0; t < 2; ++t) {
    float bv = (MODE == 1) ? bias[t * 16 + n] : 0.f;
#pragma unroll
    for (int v = 0; v < 8; ++v) {
      int gr = rowBase + lrb + v + hi;
      if (gr < NNODES) {
        size_t o = (size_t)gr * 32 + t * 16 + n;
        float val = acc[t][v];
        if (MODE == 0) {
          out0[o] = val;
          out1[o] = val * dd[v];
        } else {
          out0[o] = tanhf(val + bv);
        }
      }
    }
  }
}

// ---------------- edge scatter: agg[dst] += xw[src] * norm[e] ---------------
__global__ __launch_bounds__(256) void k_edge(const int* __restrict__ s32,
                                              const int* __restrict__ d32,
                                              const float* __restrict__ norm,
                                              const float* __restrict__ xw,
                                              float* __restrict__ agg) {
  long long tid = (long long)blockIdx.x * 256 + threadIdx.x;
  int e = (int)(tid >> 3);
  int g = (int)(tid & 7);  // 4-column group
  if (e >= NEDGES) return;
  int s = s32[e];
  int d = d32[e];
  float nrm = norm[e];
  float4 v = *(const float4*)(xw + (size_t)s * 32 + g * 4);
  float* ap = agg + (size_t)d * 32 + g * 4;
  atomicAdd(ap + 0, v.x * nrm);
  atomicAdd(ap + 1, v.y * nrm);
  atomicAdd(ap + 2, v.z * nrm);
  atomicAdd(ap + 3, v.w * nrm);
}

__global__ void k_bias_relu(const float* __restrict__ agg,
                            const float* __restrict__ b,
                            float* __restrict__ h) {
  int idx = blockIdx.x * blockDim.x + threadIdx.x;
  if (idx < NNODES * 32) h[idx] = fmaxf(agg[idx] + b[idx & 31], 0.0f);
}

// ---------------- final: out = tanh(p @ Wp2 + bp2) + priors -----------------
__global__ void k_out(const float* __restrict__ p, const float* __restrict__ Wp2,
                      const float* __restrict__ bp2,
                      const float* __restrict__ priors,
                      float* __restrict__ out) {
  int i = blockIdx.x * blockDim.x + threadIdx.x;
  if (i >= NNODES) return;
  float s = 0.f;
#pragma unroll
  for (int c = 0; c < 32; ++c) s += p[(size_t)i * 32 + c] * Wp2[c];
  out[i] = tanhf(s + bp2[0]) + priors[i];
}

extern "C" void kernel_launch(void* const* d_in, const int* in_sizes, int n_in,
                              void* d_out, int out_size, void* d_ws,
                              size_t ws_size, hipStream_t stream) {
  (void)in_sizes; (void)n_in; (void)out_size; (void)ws_size;
  const float* x = (const float*)d_in[0];
  const long long* ei = (const long long*)d_in[1];
  const float* priors = (const float*)d_in[2];
  const float* S = (const float*)d_in[3];
  const float* R = (const float*)d_in[4];
  const float* We1 = (const float*)d_in[5];
  const float* be1 = (const float*)d_in[6];
  const float* We2 = (const float*)d_in[7];
  const float* be2 = (const float*)d_in[8];
  const float* Wg[3] = {(const float*)d_in[9], (const float*)d_in[11],
                        (const float*)d_in[13]};
  const float* bg[3] = {(const float*)d_in[10], (const float*)d_in[12],
                        (const float*)d_in[14]};
  const float* Wp1 = (const float*)d_in[15];
  const float* bp1 = (const float*)d_in[16];
  const float* Wp2 = (const float*)d_in[17];
  const float* bp2 = (const float*)d_in[18];
  float* out = (float*)d_out;

  float* ws = (float*)d_ws;
  float* dis = ws;                           // N floats (deg -> dis in place)
  float* RW = dis + NNODES;                  // 256*64
  float* bufA = RW + 256 * 64;               // N*32  (h current)
  float* bufB = bufA + (size_t)NNODES * 32;  // N*32  (xw / p)
  float* bufC = bufB + (size_t)NNODES * 32;  // N*32  (agg)
  int* s32 = (int*)(bufC + (size_t)NNODES * 32);  // E ints
  int* d32 = s32 + NEDGES;                        // E ints
  float* norm = (float*)(d32 + NEDGES);           // E floats

  const int nb64 = (NNODES + 63) / 64;

  k_deg_init<<<(NNODES + 255) / 256, 256, 0, stream>>>(dis);
  k_prep_edges<<<(NEDGES + 255) / 256, 256, 0, stream>>>(ei, dis, s32, d32);
  k_dis<<<(NNODES + 255) / 256, 256, 0, stream>>>(dis);
  k_norm<<<(NEDGES + 255) / 256, 256, 0, stream>>>(s32, d32, dis, norm);
  k_rw<<<(256 * 64) / 256, 256, 0, stream>>>(R, We1, RW);
  k_embed<<<nb64, 128, 0, stream>>>(x, S, We1, be1, We2, be2, RW, bufA);
  for (int l = 0; l < 3; ++l) {
    k_gemm32<0><<<nb64, 128, 0, stream>>>(bufA, Wg[l], nullptr, dis, bufB,
                                          bufC);
    k_edge<<<(int)(((long long)NEDGES * 8 + 255) / 256), 256, 0, stream>>>(
        s32, d32, norm, bufB, bufC);
    k_bias_relu<<<(NNODES * 32 + 255) / 256, 256, 0, stream>>>(bufC, bg[l],
                                                               bufA);
  }
  k_gemm32<1><<<nb64, 128, 0, stream>>>(bufA, Wp1, bp1, nullptr, bufB, nullptr);
  k_out<<<(NNODES + 255) / 256, 256, 0, stream>>>(bufB, Wp2, bp2, priors, out);
}